// HeteroTextGCN_89309549953148
// MI455X (gfx1250) — compile-verified
//
#include <hip/hip_runtime.h>
#include <hip/hip_bf16.h>

// ---------------------------------------------------------------------------
// HeteroTextGCN for MI455X (gfx1250, wave32).
// Dense GEMMs use V_WMMA_F32_16X16X4_F32 (exact fp32 matrix path on CDNA5).
// Edge aggregation uses fp32 global atomics (L2-resident: 20MB << 192MB L2).
// Attention fused per-node in one wave32 with shfl_xor reductions.
// ---------------------------------------------------------------------------

typedef __attribute__((ext_vector_type(2))) float v2f;
typedef __attribute__((ext_vector_type(8))) float v8f;

#define GCN_N    20000
#define GCN_E    320000
#define GCN_EMB  300
#define GCN_HID  256
#define GCN_OUT  128

// ------------------------------ utility ------------------------------------

__global__ void zero_kernel(float* __restrict__ p, long long n) {
    long long i = (long long)blockIdx.x * blockDim.x + threadIdx.x;
    long long stride = (long long)gridDim.x * blockDim.x;
    for (; i < n; i += stride) p[i] = 0.0f;
}

// degO/degI: [3*N] each. Accumulate 1.0 per edge endpoint.
__global__ void degree_kernel(const int* __restrict__ s0, const int* __restrict__ d0,
                              const int* __restrict__ s1, const int* __restrict__ d1,
                              const int* __restrict__ s2, const int* __restrict__ d2,
                              float* __restrict__ degO, float* __restrict__ degI) {
    int idx = blockIdx.x * blockDim.x + threadIdx.x;
    if (idx >= 3 * GCN_E) return;
    int t = idx / GCN_E;
    int e = idx - t * GCN_E;
    const int* sp = (t == 0) ? s0 : ((t == 1) ? s1 : s2);
    const int* dp = (t == 0) ? d0 : ((t == 1) ? d1 : d2);
    atomicAdd(&degO[t * GCN_N + sp[e]], 1.0f);
    atomicAdd(&degI[t * GCN_N + dp[e]], 1.0f);
}

// in-place deg -> rsqrt(max(deg, 1))
__global__ void rsqrt_kernel(float* __restrict__ p, int n) {
    int i = blockIdx.x * blockDim.x + threadIdx.x;
    if (i < n) {
        float d = p[i];
        p[i] = rsqrtf(d < 1.0f ? 1.0f : d);
    }
}

// ------------------------------ WMMA GEMM ----------------------------------
// H[Nn,Fout] = (X[Nn,Fin] @ W[Fin,Fout]) * s[row]   (row scaling fused)
// Wave tile: 32 rows x 16 cols  (two v_wmma_f32_16x16x4_f32 accumulators
// sharing one B fragment per K-step). 8 waves / block.
// A 16x4 frag: lanes 0-15 hold M=lane, {K=k,k+1}; lanes 16-31 hold {K=k+2,k+3}.
// B 4x16 frag: N = lane&15; lanes 0-15 rows {k,k+1}, lanes 16-31 rows {k+2,k+3}.
// D:   elem r of lane -> row = r + 8*(lane>>4), col = lane&15.
__global__ void __launch_bounds__(256)
gemm_scaled_kernel(const float* __restrict__ X, const float* __restrict__ W,
                   const float* __restrict__ s, float* __restrict__ H,
                   int Nn, int Fin, int Fout) {
    const int lane   = threadIdx.x & 31;
    const int waveId = threadIdx.x >> 5;
    const int nColT  = Fout >> 4;
    const int nRowT  = Nn >> 5;                 // 32-row tiles (Nn % 32 == 0)
    const int tile   = blockIdx.x * 8 + waveId;
    if (tile >= nRowT * nColT) return;
    const int rt = tile / nColT;
    const int ct = tile - rt * nColT;
    const int r0 = rt << 5;
    const int n0 = ct << 4;
    const int m  = lane & 15;
    const int hi = lane >> 4;                   // 0 or 1 (selects K pair)

    v8f acc0 = {};
    v8f acc1 = {};
    const float* xr0 = X + (long long)(r0 + m)      * Fin;
    const float* xr1 = X + (long long)(r0 + 16 + m) * Fin;

    for (int k = 0; k < Fin; k += 4) {
        const int ka = k + 2 * hi;
        v2f a0, a1, b;
        a0.x = xr0[ka];     a0.y = xr0[ka + 1];
        a1.x = xr1[ka];     a1.y = xr1[ka + 1];
        b.x  = W[(long long)ka       * Fout + n0 + m];
        b.y  = W[(long long)(ka + 1) * Fout + n0 + m];
        acc0 = __builtin_amdgcn_wmma_f32_16x16x4_f32(
                   false, a0, false, b, (short)0, acc0, false, false);
        acc1 = __builtin_amdgcn_wmma_f32_16x16x4_f32(
                   false, a1, false, b, (short)0, acc1, false, false);
    }

    const int col = n0 + m;
#pragma unroll
    for (int r = 0; r < 8; ++r) {
        int row0 = r0 + r + hi * 8;
        int row1 = row0 + 16;
        H[(long long)row0 * Fout + col] = acc0[r] * s[row0];
        H[(long long)row1 * Fout + col] = acc1[r] * s[row1];
    }
}

// --------------------------- edge scatter-add ------------------------------
// agg[dst[e], :] += h[src[e], :]   (one thread per edge x 4 features)
__global__ void scatter_kernel(const float* __restrict__ h, const int* __restrict__ src,
                               const int* __restrict__ dst, float* __restrict__ agg,
                               int F) {
    const int fq = F >> 2;                       // float4 quads per row
    int idx = blockIdx.x * blockDim.x + threadIdx.x;
    if (idx >= GCN_E * fq) return;
    int e  = idx / fq;
    int f4 = (idx - e * fq) << 2;
    int sN = src[e];
    int dN = dst[e];
    const float4 v = *(const float4*)(h + (long long)sN * F + f4);
    float* base = agg + (long long)dN * F + f4;
    atomicAdd(base + 0, v.x);
    atomicAdd(base + 1, v.y);
    atomicAdd(base + 2, v.z);
    atomicAdd(base + 3, v.w);
}

// ------------------------- fused attention combine -------------------------
// Per node (one wave32 each):
//   h_t = agg_t[node]*sin_t[node] + b_t           (finish GraphConv)
//   d_t = h_t . a[:F] ;  e_t = h_t . a[F:]        (wave reductions)
//   sc_t = leaky0.2(d_t + mean(e)); attn = softmax_t(sc)
//   out  = sum_t attn_t * h_t ; optional leaky 0.01
template <int F, bool LEAKY01>
__global__ void __launch_bounds__(256)
attn_kernel(const float* __restrict__ agg0, const float* __restrict__ agg1,
            const float* __restrict__ agg2, const float* __restrict__ sIn,
            const float* __restrict__ b0, const float* __restrict__ b1,
            const float* __restrict__ b2, const float* __restrict__ a,
            float* __restrict__ out, int Nn) {
    constexpr int FT = F / 32;
    const int lane = threadIdx.x & 31;
    const int node = blockIdx.x * 8 + (threadIdx.x >> 5);
    if (node >= Nn) return;

    const float* aggs[3] = { agg0, agg1, agg2 };
    const float* bs[3]   = { b0, b1, b2 };

    float h[3][FT];
    float d[3], e[3];
#pragma unroll
    for (int t = 0; t < 3; ++t) {
        const float sv = sIn[t * Nn + node];
        float pd = 0.0f, pe = 0.0f;
#pragma unroll
        for (int j = 0; j < FT; ++j) {
            const int f = lane + j * 32;
            const float v = aggs[t][(long long)node * F + f] * sv + bs[t][f];
            h[t][j] = v;
            pd += v * a[f];
            pe += v * a[F + f];
        }
#pragma unroll
        for (int off = 16; off > 0; off >>= 1) {
            pd += __shfl_xor(pd, off, 32);
            pe += __shfl_xor(pe, off, 32);
        }
        d[t] = pd;
        e[t] = pe;
    }

    const float me = (e[0] + e[1] + e[2]) * (1.0f / 3.0f);
    float sc[3];
    float mx = -3.0e38f;
#pragma unroll
    for (int t = 0; t < 3; ++t) {
        float v = d[t] + me;
        sc[t] = (v > 0.0f) ? v : 0.2f * v;       // LeakyReLU(0.2)
        mx = fmaxf(mx, sc[t]);
    }
    float sum = 0.0f;
#pragma unroll
    for (int t = 0; t < 3; ++t) { sc[t] = __expf(sc[t] - mx); sum += sc[t]; }
    const float inv = 1.0f / sum;

#pragma unroll
    for (int j = 0; j < FT; ++j) {
        const int f = lane + j * 32;
        float v = (sc[0] * h[0][j] + sc[1] * h[1][j] + sc[2] * h[2][j]) * inv;
        if (LEAKY01) v = (v > 0.0f) ? v : 0.01f * v;
        out[(long long)node * F + f] = v;
    }
}

// ------------------------------- final FC ----------------------------------
__global__ void fc_kernel(const float* __restrict__ h, const float* __restrict__ w,
                          const float* __restrict__ b, float* __restrict__ logits,
                          int Nn) {
    int idx = blockIdx.x * blockDim.x + threadIdx.x;
    if (idx >= Nn * 10) return;
    int n = idx / 10;
    int c = idx - n * 10;
    float acc = b[c];
    const float* hr = h + (long long)n * GCN_OUT;
#pragma unroll 4
    for (int k = 0; k < GCN_OUT; ++k) acc += hr[k] * w[k * 10 + c];
    logits[idx] = acc;
}

// ------------------------------- launcher ----------------------------------

extern "C" void kernel_launch(void* const* d_in, const int* in_sizes, int n_in,
                              void* d_out, int out_size, void* d_ws, size_t ws_size,
                              hipStream_t stream) {
    (void)in_sizes; (void)n_in; (void)out_size; (void)ws_size;
    const float* x    = (const float*)d_in[0];
    const int*   src[3] = { (const int*)d_in[1], (const int*)d_in[3], (const int*)d_in[5] };
    const int*   dst[3] = { (const int*)d_in[2], (const int*)d_in[4], (const int*)d_in[6] };
    const float* W0[3] = { (const float*)d_in[7],  (const float*)d_in[11], (const float*)d_in[15] };
    const float* b0[3] = { (const float*)d_in[8],  (const float*)d_in[12], (const float*)d_in[16] };
    const float* W1[3] = { (const float*)d_in[9],  (const float*)d_in[13], (const float*)d_in[17] };
    const float* b1[3] = { (const float*)d_in[10], (const float*)d_in[14], (const float*)d_in[18] };
    const float* a0   = (const float*)d_in[19];
    const float* a1   = (const float*)d_in[20];
    const float* fc_w = (const float*)d_in[21];
    const float* fc_b = (const float*)d_in[22];

    const int N = GCN_N;

    // workspace carving (floats)
    float* ws   = (float*)d_ws;
    float* degO = ws;                               // 3*N  -> rsqrt(out_deg)
    float* degI = ws + 3LL * N;                     // 3*N  -> rsqrt(in_deg)
    float* agg  = ws + 6LL * N;                     // 3*N*256 (layer2 reuses 3*N*128)
    float* hsc  = agg + 3LL * N * GCN_HID;          // N*256 staging
    float* h1   = hsc + (long long)N * GCN_HID;     // N*256 layer-1 output

    float* hout   = (float*)d_out;                  // N*128 final h
    float* logits = hout + (long long)N * GCN_OUT;  // N*10

    // 1) zero degrees + layer-1 agg (contiguous region)
    {
        long long nz = 6LL * N + 3LL * N * GCN_HID;
        zero_kernel<<<4096, 256, 0, stream>>>(ws, nz);
    }
    // 2) degrees
    degree_kernel<<<(3 * GCN_E + 255) / 256, 256, 0, stream>>>(
        src[0], dst[0], src[1], dst[1], src[2], dst[2], degO, degI);
    // 3) deg -> rsqrt(max(deg,1))
    rsqrt_kernel<<<(6 * N + 255) / 256, 256, 0, stream>>>(ws, 6 * N);

    // 4) layer 1: per relation GEMM (WMMA) + scatter-add
    {
        const int tiles = (N / 32) * (GCN_HID / 16);     // 625*16 = 10000
        const int blocks = (tiles + 7) / 8;
        const int sBlocks = (GCN_E * (GCN_HID / 4) + 255) / 256;
        for (int t = 0; t < 3; ++t) {
            gemm_scaled_kernel<<<blocks, 256, 0, stream>>>(
                x, W0[t], degO + (long long)t * N, hsc, N, GCN_EMB, GCN_HID);
            scatter_kernel<<<sBlocks, 256, 0, stream>>>(
                hsc, src[t], dst[t], agg + (long long)t * N * GCN_HID, GCN_HID);
        }
    }
    // 5) attention layer 1 (+ LeakyReLU 0.01)
    attn_kernel<GCN_HID, true><<<(N + 7) / 8, 256, 0, stream>>>(
        agg, agg + (long long)N * GCN_HID, agg + 2LL * N * GCN_HID,
        degI, b0[0], b0[1], b0[2], a0, h1, N);

    // 6) zero layer-2 agg
    zero_kernel<<<4096, 256, 0, stream>>>(agg, 3LL * N * GCN_OUT);

    // 7) layer 2
    {
        const int tiles = (N / 32) * (GCN_OUT / 16);     // 625*8 = 5000
        const int blocks = (tiles + 7) / 8;
        const int sBlocks = (GCN_E * (GCN_OUT / 4) + 255) / 256;
        for (int t = 0; t < 3; ++t) {
            gemm_scaled_kernel<<<blocks, 256, 0, stream>>>(
                h1, W1[t], degO + (long long)t * N, hsc, N, GCN_HID, GCN_OUT);
            scatter_kernel<<<sBlocks, 256, 0, stream>>>(
                hsc, src[t], dst[t], agg + (long long)t * N * GCN_OUT, GCN_OUT);
        }
    }
    // 8) attention layer 2 -> final h straight into d_out
    attn_kernel<GCN_OUT, false><<<(N + 7) / 8, 256, 0, stream>>>(
        agg, agg + (long long)N * GCN_OUT, agg + 2LL * N * GCN_OUT,
        degI, b1[0], b1[1], b1[2], a1, hout, N);

    // 9) logits
    fc_kernel<<<(N * 10 + 255) / 256, 256, 0, stream>>>(hout, fc_w, fc_b, logits, N);
}